// DiriAdaptiveLabelLoss_69346541961875
// MI455X (gfx1250) — compile-verified
//
#include <hip/hip_runtime.h>

// ---------------------------------------------------------------------------
// DiriAdaptiveLabelLoss for MI455X (gfx1250, wave32)
//
// loss_b = logsumexp(pred_b) - ( 0.9*pred_b[t] + 0.1/csum_t * sum_j w_t[j]*pred_b[j] )
//   w_t[j] = (j==t) ? 0 : conf[t][j - (j>t)],  csum_t = sum(conf[t][:])
//
//  * one 256-thread block (8 waves) per 16-row tile (2048 blocks)
//  * weight rows pre-expanded (shift + zero-at-target) into LDS -> branchless hot loop
//  * pred streamed via GLOBAL_LOAD_ASYNC_TO_LDS_B128, double-buffered K-panels,
//    s_wait_asynccnt pipeline (ASYNCcnt), coalesced 16B/lane
//  * row reductions (sum exp, weighted dot) via V_WMMA_F32_16X16X4_F32 with a
//    ones B-matrix: D[m,n] += sum_k A[m,k]  (f32 matrix-pipe accumulation)
//  * per-row losses -> workspace; deterministic tree reduction -> scalar mean
// ---------------------------------------------------------------------------

typedef __attribute__((ext_vector_type(2))) float v2f;
typedef __attribute__((ext_vector_type(8))) float v8f;

#define C_DIM   1000
#define CM1     999
#define WSTRIDE 1004   // 1004 % 64 == 44 -> 44*r mod 64 distinct for r=0..15
#define PANEL_K 256    // floats per panel per row
#define PSTRIDE 260    // 260 % 64 == 4  -> conflict-free WMMA feed
#define NPANEL  4      // 256+256+256+232 = 1000

__device__ __forceinline__ void async_copy_b128(unsigned lds_off, const float* g) {
  asm volatile("global_load_async_to_lds_b128 %0, %1, off"
               :: "v"(lds_off), "v"(g) : "memory");
}

__global__ __launch_bounds__(256) void diri_loss_rows_kernel(
    const float* __restrict__ pred,     // [B, 1000]
    const int* __restrict__ target,     // [B]
    const float* __restrict__ conf,     // [1000, 999]
    float* __restrict__ rowloss)        // [B]
{
  __shared__ __align__(16) float wtile[16 * WSTRIDE];      // expanded weights (64256 B)
  __shared__ __align__(16) float ptile[2][16 * PSTRIDE];   // pred panels (2 x 16640 B)
  __shared__ float csum[16];
  __shared__ float predt[16];
  __shared__ int   tgt[16];
  __shared__ float partE[8][16];
  __shared__ float partW[8][16];

  const int tid  = threadIdx.x;
  const int lane = tid & 31;
  const int wv   = tid >> 5;
  const int r0   = blockIdx.x * 16;

  // --- targets + pred[b, t] ----------------------------------------------
  if (tid < 16) {
    int t = target[r0 + tid];
    tgt[tid] = t;
    predt[tid] = pred[(size_t)(r0 + tid) * C_DIM + t];
  }
  __syncthreads();

  // --- kick off async staging of pred panel 0 (overlaps weight expansion) --
  auto stage = [&](int p) {
    float* buf = ptile[p & 1];
    const unsigned lbase = (unsigned)(uintptr_t)buf;   // low 32 bits = LDS offset
    const int k0   = p * PANEL_K;
    const int cmax = (p == NPANEL - 1) ? 57 : 63;      // 58 or 64 b128 chunks/row
    #pragma unroll
    for (int i = 0; i < 4; ++i) {                      // exactly 4 asyncs per thread
      int idx = tid + (i << 8);                        // 0..1023
      int row = idx >> 6;
      int c   = idx & 63;
      if (c > cmax) c = cmax;                          // clamp: dup writes, same data
      const float* g = pred + (size_t)(r0 + row) * C_DIM + k0 + (c << 2);
      unsigned l = lbase + (unsigned)(row * (PSTRIDE * 4) + (c << 4));
      async_copy_b128(l, g);
    }
  };
  stage(0);

  // --- expand confusion rows into branchless weight vectors + row sums -----
  // wave w handles tile-rows 2w, 2w+1
  for (int rr = wv * 2; rr < wv * 2 + 2; ++rr) {
    const int t = tgt[rr];
    const float* crow_g = conf + (size_t)t * CM1;
    float s = 0.f;
    for (int j = lane; j < C_DIM; j += 32) {
      int sidx = j - (j > t ? 1 : 0);
      if (sidx > CM1 - 1) sidx = CM1 - 1;              // keep load in-bounds
      float v = crow_g[sidx];                          // unconditional load
      v = (j == t) ? 0.f : v;                          // branchless zero insert
      wtile[rr * WSTRIDE + j] = v;
      s += v;
    }
    for (int off = 16; off > 0; off >>= 1) s += __shfl_xor(s, off, 32);
    if (lane == 0) csum[rr] = s;
  }

  // --- WMMA feed mapping (A 16x4 f32): lane<16 -> A[L,0..1], lane>=16 -> A[L,2..3]
  const int arow = lane & 15;
  const int koff = (lane >> 4) << 1;
  const float* wrow = &wtile[arow * WSTRIDE];

  v2f ones; ones.x = 1.f; ones.y = 1.f;
  v8f accE = {0.f, 0.f, 0.f, 0.f, 0.f, 0.f, 0.f, 0.f};
  v8f accW = {0.f, 0.f, 0.f, 0.f, 0.f, 0.f, 0.f, 0.f};

  // --- double-buffered panel pipeline --------------------------------------
  for (int p = 0; p < NPANEL; ++p) {
    if (p + 1 < NPANEL) {
      stage(p + 1);                                   // async DMA next panel
      asm volatile("s_wait_asynccnt 0x4" ::: "memory");  // panel p landed
    } else {
      asm volatile("s_wait_asynccnt 0x0" ::: "memory");
    }
    __syncthreads();                                  // all waves' stage of p done

    const int k0     = p * PANEL_K;
    const int nchunk = (p == NPANEL - 1) ? 58 : 64;   // K/4 chunks this panel
    const float* lrow = &ptile[p & 1][arow * PSTRIDE];

    #pragma unroll 2
    for (int c = wv; c < nchunk; c += 8) {            // wave-uniform: EXEC all 1s
      const int j = (c << 2) + koff;
      v2f pv = *(const v2f*)(lrow + j);               // ds_load_b64 (conflict-free)
      v2f wp = *(const v2f*)(wrow + k0 + j);          // ds_load_b64 (branchless wts)

      v2f aE; aE.x = __expf(pv.x); aE.y = __expf(pv.y);
      accE = __builtin_amdgcn_wmma_f32_16x16x4_f32(
          false, aE, false, ones, (short)0, accE, false, false);

      v2f aW; aW.x = wp.x * pv.x; aW.y = wp.y * pv.y;
      accW = __builtin_amdgcn_wmma_f32_16x16x4_f32(
          false, aW, false, ones, (short)0, accW, false, false);
    }
    __syncthreads();                                  // reads done before reuse
  }

  // D layout: VGPR r / lane 0 -> row r; VGPR r / lane 16 -> row r+8.
  if (lane == 0) {
    #pragma unroll
    for (int r = 0; r < 8; ++r) { partE[wv][r] = accE[r]; partW[wv][r] = accW[r]; }
  } else if (lane == 16) {
    #pragma unroll
    for (int r = 0; r < 8; ++r) { partE[wv][r + 8] = accE[r]; partW[wv][r + 8] = accW[r]; }
  }
  __syncthreads();

  // --- combine wave partials, finalize per-row loss ------------------------
  if (tid < 16) {
    float sE = 0.f, sW = 0.f;
    #pragma unroll
    for (int w = 0; w < 8; ++w) { sE += partE[w][tid]; sW += partW[w][tid]; }
    const float lse  = __logf(sE);
    const float loss = lse - (0.9f * predt[tid] + 0.1f * sW / csum[tid]);
    rowloss[r0 + tid] = loss;
  }
}

__global__ __launch_bounds__(256) void diri_loss_mean_kernel(
    const float* __restrict__ rowloss, float* __restrict__ out, int n)
{
  __shared__ float red[256];
  float s = 0.f;
  for (int i = threadIdx.x; i < n; i += 256) s += rowloss[i];
  red[threadIdx.x] = s;
  __syncthreads();
  for (int off = 128; off > 0; off >>= 1) {
    if ((int)threadIdx.x < off) red[threadIdx.x] += red[threadIdx.x + off];
    __syncthreads();
  }
  if (threadIdx.x == 0) out[0] = red[0] / (float)n;
}

extern "C" void kernel_launch(void* const* d_in, const int* in_sizes, int n_in,
                              void* d_out, int out_size, void* d_ws, size_t ws_size,
                              hipStream_t stream) {
  const float* pred   = (const float*)d_in[0];
  const int*   target = (const int*)d_in[1];
  const float* conf   = (const float*)d_in[2];
  float* rowloss = (float*)d_ws;
  float* out     = (float*)d_out;

  const int B = in_sizes[1];          // 32768
  const int blocks = B / 16;          // 2048 tiles of 16 rows

  diri_loss_rows_kernel<<<blocks, 256, 0, stream>>>(pred, target, conf, rowloss);
  diri_loss_mean_kernel<<<1, 256, 0, stream>>>(rowloss, out, B);
}